// Reduce_30726196036189
// MI455X (gfx1250) — compile-verified
//
#include <hip/hip_runtime.h>

// Scatter-add: out[b, tgt[b,e], :] += messages[b, e, :]
// B=32, E=8192, N=2048, D=128, fp32.
// Memory-bound: ~170 MB traffic -> ~7 us floor at 23.3 TB/s.
//
// CDNA5 strategy:
//   * One TDM DMA (tensor_load_to_lds) per 256-thread block stages a
//     contiguous 64-edge x 512 B = 32 KB message tile into LDS
//     (TENSORcnt-tracked, no VGPR traffic, single instruction).
//   * Waves then spend all vector-memory issue slots on the atomics:
//     ds_load_b128 from the LDS tile + 4x global_atomic_add_f32 per lane
//     into the L2-resident output (33.5 MB << 192 MB L2).

typedef float        v4f __attribute__((ext_vector_type(4)));
typedef unsigned int v4u __attribute__((ext_vector_type(4)));
typedef int          v4i __attribute__((ext_vector_type(4)));
typedef int          v8i __attribute__((ext_vector_type(8)));

#define B_ 32
#define E_ 8192
#define N_ 2048
#define D_ 128

#define EDGES_PER_BLOCK 64
#define THREADS 256
#define WAVES_PER_BLOCK (THREADS / 32)
#define EDGES_PER_WAVE (EDGES_PER_BLOCK / WAVES_PER_BLOCK)   // 8
#define TILE_ELTS (EDGES_PER_BLOCK * D_)                     // 8192 f32 = 32 KB

__global__ void zero_out_kernel(v4f* __restrict__ out) {
    size_t i = (size_t)blockIdx.x * blockDim.x + threadIdx.x;
    v4f z = {0.f, 0.f, 0.f, 0.f};
    out[i] = z;   // global_store_b128
}

__device__ __forceinline__ void gatomic_add_f32(float* addr, float v) {
    // No-return f32 atomic add, device scope (L2 atomic units).
    asm volatile("global_atomic_add_f32 %0, %1, off scope:SCOPE_DEV"
                 :: "v"(addr), "v"(v)
                 : "memory");
}

__global__ void scatter_add_kernel(const float* __restrict__ messages,
                                   const int*   __restrict__ tgt,
                                   float*       __restrict__ out) {
    __shared__ __attribute__((aligned(16))) float tile[TILE_ELTS];

    const int blockEdge0 = blockIdx.x * EDGES_PER_BLOCK;

    // ---- Wave 0: build D# and fire the TDM bulk copy (EXEC-independent) ----
    if (threadIdx.x < 32) {
        unsigned long long gaddr =
            (unsigned long long)(const void*)messages +
            (unsigned long long)blockEdge0 * (D_ * 4);   // byte addr of tile
        unsigned int lds_addr = (unsigned int)(unsigned long long)
            (__attribute__((address_space(3))) float*)tile;

        // D# group 0 (128b): count=1 | lds_addr | global_addr[56:0] | type=2
        v4u g0;
        g0.x = 1u;                                        // count=1, user mode
        g0.y = lds_addr;                                  // LDS byte address
        g0.z = (unsigned int)(gaddr & 0xffffffffull);     // global_addr[31:0]
        g0.w = (unsigned int)((gaddr >> 32) & 0x01ffffffull)
             | (2u << 30);                                // addr[56:32] | type=2

        // D# group 1 (256b): data_size=4B, 1 row of TILE_ELTS elements.
        v8i g1;
        g1[0] = (int)(2u << 16);                 // workgroup_mask=0, data_size=2 (4B)
        g1[1] = (int)((TILE_ELTS & 0xffffu) << 16);      // tensor_dim0[15:0]
        g1[2] = (int)((TILE_ELTS >> 16) | (1u << 16));   // tensor_dim0[31:16] | tensor_dim1=1
        g1[3] = (int)((TILE_ELTS & 0xffffu) << 16);      // tile_dim0 = TILE_ELTS
        g1[4] = 1;                                       // tile_dim1 = 1, tile_dim2 = 0
        g1[5] = TILE_ELTS;                               // tensor_dim0_stride[31:0]
        g1[6] = 0;                                       // stride[47:32] | dim1_stride lo
        g1[7] = 0;                                       // dim1_stride hi
        v4i g2 = {0, 0, 0, 0};                           // <=2D: zero-filled
        v4i g3 = {0, 0, 0, 0};
        v8i g4 = {0, 0, 0, 0, 0, 0, 0, 0};               // 6-arg form extra group

        __builtin_amdgcn_tensor_load_to_lds(g0, g1, g2, g3, g4, /*cpol=*/0);
        __builtin_amdgcn_s_wait_tensorcnt(0);
    }
    __syncthreads();   // tile visible to all 8 waves

    // ---- All waves: drain LDS tile into output via L2 atomics ----
    const int lane = threadIdx.x & 31;
    const int wave = threadIdx.x >> 5;

    for (int j = 0; j < EDGES_PER_WAVE; ++j) {
        const int le = wave * EDGES_PER_WAVE + j;        // edge within tile
        const int g  = blockEdge0 + le;                  // global edge id
        const int b  = g >> 13;                          // g / E_
        const int t  = tgt[g];                           // uniform per wave

        const v4f m = *reinterpret_cast<const v4f*>(
            tile + le * D_ + (lane << 2));               // ds_load_b128

        float* o = out + ((size_t)(b * N_ + t) * D_ + (size_t)(lane << 2));
        gatomic_add_f32(o + 0, m.x);
        gatomic_add_f32(o + 1, m.y);
        gatomic_add_f32(o + 2, m.z);
        gatomic_add_f32(o + 3, m.w);
    }
}

extern "C" void kernel_launch(void* const* d_in, const int* in_sizes, int n_in,
                              void* d_out, int out_size, void* d_ws, size_t ws_size,
                              hipStream_t stream) {
    const float* messages = (const float*)d_in[0];   // (B, E, D) f32
    const int*   tgt      = (const int*)d_in[1];     // (B, E)    i32
    // d_in[2] (atom_features_ref) only defines the output shape - unused.
    float* out = (float*)d_out;                      // (B, N, D) f32

    // 1) Zero the (poisoned) output: 8,388,608 floats = 2,097,152 float4s.
    const int n4 = out_size / 4;
    zero_out_kernel<<<n4 / 256, 256, 0, stream>>>((v4f*)out);

    // 2) Scatter: 262,144 edges, 64 edges (one 32 KB TDM tile) per block.
    const int edges = B_ * E_;
    scatter_add_kernel<<<edges / EDGES_PER_BLOCK, THREADS, 0, stream>>>(
        messages, tgt, out);
}